// EncoderLayer_66279935312082
// MI455X (gfx1250) — compile-verified
//
#include <hip/hip_runtime.h>

typedef __attribute__((ext_vector_type(2))) float v2f;
typedef __attribute__((ext_vector_type(8))) float v8f;

#define D 128
#define BN_EPS 1e-5f

// ---------------------------------------------------------------------------
// Zero the workspace: h accumulator [n_nodes*128] + column sum/sumsq [2*128]
// ---------------------------------------------------------------------------
__global__ void zero_ws_kernel(float* __restrict__ ws, long n) {
  long i = (long)blockIdx.x * blockDim.x + threadIdx.x;
  long stride = (long)gridDim.x * blockDim.x;
  for (; i < n; i += stride) ws[i] = 0.0f;
}

// ---------------------------------------------------------------------------
// Edge scatter-add: h[dst[e]] += x[src[e]].  One wave32 per edge: 32 lanes x
// float4 = the full 128-float row, fully coalesced 512B read + atomic adds.
// x (25.6MB) and h (25.6MB) are both L2-resident (192MB L2) -> L2 atomics.
// ---------------------------------------------------------------------------
__global__ void scatter_kernel(const float* __restrict__ x,
                               const int* __restrict__ src,
                               const int* __restrict__ dst,
                               float* __restrict__ h, int n_edges) {
  int lane = threadIdx.x & 31;
  int wid  = blockIdx.x * (blockDim.x >> 5) + (threadIdx.x >> 5);
  int nw   = gridDim.x * (blockDim.x >> 5);
  for (int e = wid; e < n_edges; e += nw) {
    int s = src[e];             // uniform per wave -> scalar load
    int d = dst[e];
    float4 v = ((const float4*)(x + (size_t)s * D))[lane];
    float* hp = h + (size_t)d * D + lane * 4;
    atomicAdd(hp + 0, v.x);
    atomicAdd(hp + 1, v.y);
    atomicAdd(hp + 2, v.z);
    atomicAdd(hp + 3, v.w);
  }
}

// ---------------------------------------------------------------------------
// WMMA f32 GEMM: out = relu(h @ W + b), fused column-stat accumulation for BN.
// Each wave computes 16 rows x 128 cols using V_WMMA_F32_16X16X4_F32:
//   A 16x4  : lane<16 -> (M=lane, K=k..k+1), lane>=16 -> (M=lane-16, K=k+2..k+3)
//   B 4x16  : lane<16 -> (K=k..k+1, N=lane), lane>=16 -> (K=k+2..k+3, N=lane-16)
//   C 16x16 : N = lane%16, M = vgpr + 8*(lane/16)
// W is staged in LDS pre-packed as float2 pairs {W[2kp][n], W[2kp+1][n]} so a
// single ds_load_b64 delivers the exact even-aligned WMMA B operand pair
// (eliminates the v_mov repacking seen when the compiler fuses loads across
// N-tiles).  A fragments are double-buffered to hide global load latency.
// ---------------------------------------------------------------------------
__global__ void __launch_bounds__(256)
gemm_relu_stats_kernel(const float* __restrict__ h, const float* __restrict__ W,
                       const float* __restrict__ bias, float* __restrict__ out,
                       float* __restrict__ colsum, float* __restrict__ colsumsq,
                       int n_nodes) {
  __shared__ v2f lds_b2[(D / 2) * D];  // [kp][n] = {W[2kp][n], W[2kp+1][n]}, 64KB

  // Cooperative packed-transpose load of W into LDS (coalesced b32 reads).
  for (int i = threadIdx.x; i < (D / 2) * D; i += blockDim.x) {
    int kp = i >> 7;      // i / 128
    int n  = i & (D - 1); // i % 128
    v2f v;
    v.x = W[(2 * kp) * D + n];
    v.y = W[(2 * kp + 1) * D + n];
    lds_b2[i] = v;
  }
  __syncthreads();

  int lane = threadIdx.x & 31;
  int wid  = blockIdx.x * (blockDim.x >> 5) + (threadIdx.x >> 5);
  int row0 = wid * 16;
  if (row0 >= n_nodes) return;  // uniform per wave; after the only barrier

  int nl   = lane & 15;   // N (and A-row M) index within tile
  int hi   = lane >> 4;   // half-wave select
  int koff = hi * 2;      // K offset carried by upper half-wave

  v8f acc[8] = {};        // 8 N-tiles of 16 cols -> full 128-wide output block

  const float* arow = h + (size_t)(row0 + nl) * D + koff;

  v2f a_cur = *(const v2f*)(arow);  // K-chunk 0 (8B-aligned global_load_b64)
  for (int k = 0; k < D; k += 4) {
    // Branch-free prefetch of the next A fragment (wraps to 0 on last iter).
    v2f a_nxt = *(const v2f*)(arow + ((k + 4) & (D - 1)));
    const v2f* brow = lds_b2 + ((k >> 1) + hi) * D + nl;
#pragma unroll
    for (int t = 0; t < 8; ++t) {
      v2f bf = brow[t * 16];  // single ds_load_b64 -> aligned operand pair
      acc[t] = __builtin_amdgcn_wmma_f32_16x16x4_f32(
          /*neg_a=*/false, a_cur, /*neg_b=*/false, bf,
          /*c_mod=*/(short)0, acc[t], /*reuse_a=*/false, /*reuse_b=*/false);
    }
    a_cur = a_nxt;
  }

  // Epilogue: bias + ReLU, write out, accumulate per-column BN statistics.
  int mbase = row0 + hi * 8;
#pragma unroll
  for (int t = 0; t < 8; ++t) {
    int n = t * 16 + nl;
    float bb = bias[n];
    float s = 0.0f, s2 = 0.0f;
#pragma unroll
    for (int i = 0; i < 8; ++i) {
      float v = acc[t][i] + bb;
      v = fmaxf(v, 0.0f);
      out[(size_t)(mbase + i) * D + n] = v;
      s  += v;
      s2 += v * v;
    }
    atomicAdd(&colsum[n], s);
    atomicAdd(&colsumsq[n], s2);
  }
}

// ---------------------------------------------------------------------------
// Fold column stats into affine scale/shift: y = y*scale[c] + shift[c]
// ---------------------------------------------------------------------------
__global__ void bn_prepare_kernel(const float* __restrict__ colsum,
                                  const float* __restrict__ colsumsq,
                                  const float* __restrict__ gamma,
                                  const float* __restrict__ beta,
                                  float* __restrict__ scale,
                                  float* __restrict__ shift, int n_nodes) {
  int c = threadIdx.x;
  if (c < D) {
    float inv_n = 1.0f / (float)n_nodes;
    float mean = colsum[c] * inv_n;
    float var  = colsumsq[c] * inv_n - mean * mean;  // biased variance
    float rs = rsqrtf(var + BN_EPS);
    float sc = gamma[c] * rs;
    scale[c] = sc;
    shift[c] = beta[c] - mean * sc;
  }
}

__global__ void bn_apply_kernel(float* __restrict__ out,
                                const float* __restrict__ scale,
                                const float* __restrict__ shift, long total) {
  long i = (long)blockIdx.x * blockDim.x + threadIdx.x;
  long stride = (long)gridDim.x * blockDim.x;
  for (; i < total; i += stride) {
    int c = (int)(i & (D - 1));
    out[i] = out[i] * scale[c] + shift[c];
  }
}

// ---------------------------------------------------------------------------
extern "C" void kernel_launch(void* const* d_in, const int* in_sizes, int n_in,
                              void* d_out, int out_size, void* d_ws, size_t ws_size,
                              hipStream_t stream) {
  const float* x     = (const float*)d_in[0];
  const int*   src   = (const int*)d_in[1];
  const int*   dst   = (const int*)d_in[2];
  const float* W     = (const float*)d_in[3];
  const float* bias  = (const float*)d_in[4];
  const float* gamma = (const float*)d_in[5];
  const float* beta  = (const float*)d_in[6];
  float* out = (float*)d_out;

  int n_nodes = in_sizes[0] / D;
  int n_edges = in_sizes[1];

  float* ws       = (float*)d_ws;
  float* h        = ws;                          // [n_nodes * D]
  float* colsum   = ws + (size_t)n_nodes * D;    // [D]
  float* colsumsq = colsum + D;                  // [D]
  float* scale    = colsumsq + D;                // [D]
  float* shift    = scale + D;                   // [D]

  long zero_n = (long)n_nodes * D + 2 * D;
  zero_ws_kernel<<<1024, 256, 0, stream>>>(ws, zero_n);

  scatter_kernel<<<4096, 256, 0, stream>>>(x, src, dst, h, n_edges);

  int tiles   = (n_nodes + 15) / 16;             // 16-row tiles (50000 -> 3125)
  int nblocks = (tiles + 7) / 8;                 // 8 waves per 256-thread block
  gemm_relu_stats_kernel<<<nblocks, 256, 0, stream>>>(h, W, bias, out,
                                                      colsum, colsumsq, n_nodes);

  bn_prepare_kernel<<<1, 128, 0, stream>>>(colsum, colsumsq, gamma, beta,
                                           scale, shift, n_nodes);

  bn_apply_kernel<<<1024, 256, 0, stream>>>(out, scale, shift,
                                            (long)n_nodes * D);
}